// MixedScaleDenseLayer_46901042872465
// MI455X (gfx1250) — compile-verified
//
#include <hip/hip_runtime.h>

typedef float    v8f  __attribute__((ext_vector_type(8)));
typedef float    v2f  __attribute__((ext_vector_type(2)));
typedef _Float16 h8   __attribute__((ext_vector_type(8)));
typedef unsigned int u32x4 __attribute__((ext_vector_type(4)));
typedef int          i32x4 __attribute__((ext_vector_type(4)));
typedef int          i32x8 __attribute__((ext_vector_type(8)));

#define TILE_H 24
#define TILE_W 16
#define HALO   10
#define EXT_H  44           // TILE_H + 2*HALO
#define EXT_W  36           // TILE_W + 2*HALO
#define NPX    (EXT_H*EXT_W)   // 1584 extended pixels
#define NGRP   (NPX/16)        // 99 pixel-groups of 16
#define NPROJ  96              // 90 (j,t) projections, padded to 6x16
#define CIN    32
#define WIMG   512
#define HIMG   512
#define HW     (WIMG*HIMG)
#define NDIL   10
#define COUT   42
#define BLOCK  384             // 12 waves
#define NWAVE  12
#define NWELEM (NDIL*CIN*9)    // 2880 weight elements
#define WRE_BYTES (NPROJ*CIN*4)    // 12288

// Fused mixed-scale-dense layer:
//   Stage A: U[(j,t), p] = sum_c W[j,c,t] * x[c, p]  over the haloed tile,
//            via V_WMMA_F32_16X16X4_F32 (exact f32 math); U stored fp16 in LDS.
//   Stage B: out[j, p] = b[j] + sum_t U[(j,t), p + d_j * delta_t]  (LDS gather)
//            plus passthrough copy of the 32 input channels.
// Weights arrive in LDS via the Tensor Data Mover (tensor_load_to_lds),
// staged into the (not-yet-written) U region and reclaimed after reorg.
__global__ __launch_bounds__(BLOCK, 1)
void msd_fused(const float* __restrict__ x, const float* __restrict__ Wg,
               const float* __restrict__ bias, float* __restrict__ out)
{
  extern __shared__ char smem[];
  float*    Wre    = reinterpret_cast<float*>(smem);                 // 96x32 f32
  _Float16* U      = reinterpret_cast<_Float16*>(smem + WRE_BYTES);  // 1584x96 f16
  float*    Wstage = reinterpret_cast<float*>(smem + WRE_BYTES);     // aliases U start

  const int tid  = threadIdx.x;
  const int lane = tid & 31;
  const int wv   = tid >> 5;     // 12 waves
  const int lh   = lane >> 4;    // half-wave: K-pair select (A/B), M-high (C)
  const int li   = lane & 15;    // M index (A) / N index (B, C)

  const int h0 = (int)blockIdx.y * TILE_H - HALO;
  const int w0 = (int)blockIdx.x * TILE_W - HALO;
  const size_t bimg = (size_t)blockIdx.z;
  const float* xb = x + bimg * (size_t)CIN * HW;

  // ---- TDM: DMA the raw 11.5KB weight block into LDS staging ----
#if __has_builtin(__builtin_amdgcn_tensor_load_to_lds) && __has_builtin(__builtin_amdgcn_s_wait_tensorcnt)
  if (wv == 0) {
    const unsigned long long ga = (unsigned long long)(uintptr_t)Wg;
    const unsigned int lds_off  = (unsigned int)(uintptr_t)Wstage;   // low 32 bits = LDS byte offset
    u32x4 g0;
    g0[0] = 1u;                                   // count=1 (valid user descriptor)
    g0[1] = lds_off;                              // lds_addr
    g0[2] = (unsigned int)ga;                     // global_addr[31:0]
    g0[3] = (unsigned int)((ga >> 32) & 0x01FFFFFFu) | 0x80000000u;  // addr[56:32] | type=2
    i32x8 g1;
    g1[0] = (int)0x00020000u;                     // data_size = 4B
    g1[1] = (int)((unsigned)NWELEM << 16);        // tensor_dim0[15:0] = 2880
    g1[2] = (int)(1u << 16);                      // tensor_dim0 hi = 0, tensor_dim1 = 1
    g1[3] = (int)((unsigned)NWELEM << 16);        // tensor_dim1 hi = 0, tile_dim0 = 2880
    g1[4] = 1;                                    // tile_dim1 = 1, tile_dim2 = 0
    g1[5] = NWELEM;                               // tensor_dim0_stride
    g1[6] = 0;
    g1[7] = 0;
    i32x4 z4 = {0, 0, 0, 0};                      // groups 2/3 unused (<=2D tensor)
    i32x8 z8 = {0, 0, 0, 0, 0, 0, 0, 0};          // trailing group (6-arg toolchain form)
    __builtin_amdgcn_tensor_load_to_lds(g0, g1, z4, z4, z8, 0);
    __builtin_amdgcn_s_wait_tensorcnt(0);
  }
  __syncthreads();
  const float* Wsrc = Wstage;
#else
  const float* Wsrc = Wg;
#endif

  // Reorganize weights: Wre[m][c] = W[j][c][t], m = j*9 + t  (rows 90..95 zero)
  for (int i = tid; i < NPROJ * CIN; i += BLOCK) {
    int m = i >> 5, c = i & 31;
    float v = 0.0f;
    if (m < 90) v = Wsrc[(m / 9) * (CIN * 9) + c * 9 + (m % 9)];
    Wre[i] = v;
  }
  __syncthreads();   // staging area may now be overwritten by U

  // ---------------- Stage A: projection GEMM (f32 WMMA) ----------------
  for (int g = wv; g < NGRP; g += NWAVE) {
    const int p  = g * 16 + li;           // this lane's extended-pixel column
    const int er = p / EXT_W, ec = p % EXT_W;
    const int hh = h0 + er, ww = w0 + ec;
    const bool inb = (hh >= 0) && (hh < HIMG) && (ww >= 0) && (ww < WIMG);
    // clamp address so speculative loads stay in-bounds; value select -> 0
    const float* xp = xb + (size_t)(inb ? hh : 0) * WIMG + (inb ? ww : 0);

    v8f acc0 = {}, acc1 = {}, acc2 = {}, acc3 = {}, acc4 = {}, acc5 = {};
#pragma unroll
    for (int ks = 0; ks < 8; ++ks) {
      const int k = ks * 4 + lh * 2;      // this lane's K pair
      v2f bm;
      bm.x = inb ? xp[(size_t)k * HW]       : 0.0f;
      bm.y = inb ? xp[(size_t)(k + 1) * HW] : 0.0f;
#define WMMA_STEP(MT, ACC)                                                     \
      {                                                                        \
        v2f am = *reinterpret_cast<const v2f*>(&Wre[((MT) * 16 + li) * CIN + k]); \
        ACC = __builtin_amdgcn_wmma_f32_16x16x4_f32(                           \
            false, am, false, bm, (short)0, ACC, false, false);                \
      }
      WMMA_STEP(0, acc0) WMMA_STEP(1, acc1) WMMA_STEP(2, acc2)
      WMMA_STEP(3, acc3) WMMA_STEP(4, acc4) WMMA_STEP(5, acc5)
#undef WMMA_STEP
    }

    // C layout: vgpr r, lanes<16 -> M=r, lanes>=16 -> M=r+8 ; N = li.
    // The 8 rows per lane are contiguous proj indices -> one 16B fp16 store.
#define STORE_U(MT, ACC)                                                       \
    {                                                                          \
      h8 hv;                                                                   \
      hv[0] = (_Float16)ACC[0]; hv[1] = (_Float16)ACC[1];                      \
      hv[2] = (_Float16)ACC[2]; hv[3] = (_Float16)ACC[3];                      \
      hv[4] = (_Float16)ACC[4]; hv[5] = (_Float16)ACC[5];                      \
      hv[6] = (_Float16)ACC[6]; hv[7] = (_Float16)ACC[7];                      \
      *reinterpret_cast<h8*>(&U[(size_t)p * NPROJ + (MT) * 16 + lh * 8]) = hv; \
    }
    STORE_U(0, acc0) STORE_U(1, acc1) STORE_U(2, acc2)
    STORE_U(3, acc3) STORE_U(4, acc4) STORE_U(5, acc5)
#undef STORE_U
  }
  __syncthreads();

  // ---------------- Stage B: gather + bias + passthrough ----------------
  const int pr = tid >> 4, pc = tid & 15;          // one thread per output pixel
  const int h = (int)blockIdx.y * TILE_H + pr;
  const int w = (int)blockIdx.x * TILE_W + pc;
  if (h < HIMG) {                                  // last row-tile is partial
    const size_t pix = (size_t)h * WIMG + w;
    float* ob = out + bimg * (size_t)COUT * HW + pix;
    const float* xi = xb + pix;

#pragma unroll 4
    for (int c = 0; c < CIN; ++c)
      ob[(size_t)c * HW] = xi[(size_t)c * HW];

#pragma unroll
    for (int j = 0; j < NDIL; ++j) {
      const int d = j + 1;
      float acc = bias[j];
      int t = 0;
#pragma unroll
      for (int dh = -1; dh <= 1; ++dh)
#pragma unroll
        for (int dw = -1; dw <= 1; ++dw, ++t) {
          const int er = pr + HALO + dh * d;       // always in [0, EXT_H)
          const int ec = pc + HALO + dw * d;       // always in [0, EXT_W)
          acc += (float)U[(size_t)(er * EXT_W + ec) * NPROJ + j * 9 + t];
        }
      ob[(size_t)(32 + j) * HW] = acc;
    }
  }
}

extern "C" void kernel_launch(void* const* d_in, const int* in_sizes, int n_in,
                              void* d_out, int out_size, void* d_ws, size_t ws_size,
                              hipStream_t stream) {
  (void)n_in; (void)out_size; (void)d_ws; (void)ws_size;
  const float* x = (const float*)d_in[0];
  const float* W = (const float*)d_in[1];
  const float* b = (const float*)d_in[2];
  float* out = (float*)d_out;

  const int nbatch = in_sizes[0] / (CIN * HW);     // 8
  const size_t shmem = WRE_BYTES + (size_t)NPX * NPROJ * sizeof(_Float16);  // 316,416 B

  // Raise the dynamic-LDS cap toward CDNA5's 320KB-per-workgroup limit (idempotent).
  (void)hipFuncSetAttribute(reinterpret_cast<const void*>(msd_fused),
                            hipFuncAttributeMaxDynamicSharedMemorySize, (int)shmem);

  dim3 grid(WIMG / TILE_W, (HIMG + TILE_H - 1) / TILE_H, nbatch);  // 32 x 22 x 8
  msd_fused<<<grid, BLOCK, shmem, stream>>>(x, W, b, out);
}